// BlurModel_1520418423329
// MI455X (gfx1250) — compile-verified
//
#include <hip/hip_runtime.h>
#include <math.h>

typedef __attribute__((ext_vector_type(2))) float v2f;
typedef __attribute__((ext_vector_type(4))) float v4f;
typedef __attribute__((ext_vector_type(8))) float v8f;

#define IMG_W 2048
#define IMG_H 2048
#define TILES_X (IMG_W / 16)          // 128
#define NTILES  (TILES_X * TILES_X)   // 16384
#define WAVES_PER_BLOCK 8

// ---------------------------------------------------------------------------
// 1) Horizontal 9-tap box sum via V_WMMA_F32_16X16X4_F32:
//    O(16x16) = I(16x24) x Band(24x16),  Band[j][n] = 1 if j in [n, n+8]
//    One wave per 16x16 tile. A: lane m=l%16, K = 2*(l/16)+v.
//    B/C/D: row per VGPR, lane-half adds rows/2.
//    Boundary: pair start column is even, invalid ranges are [-4,-1] and
//    [2048,2051] -> a b64 pair is uniformly valid/invalid: clamp + select.
// ---------------------------------------------------------------------------
__global__ __launch_bounds__(256) void blur_h_wmma(const float* __restrict__ x,
                                                   float* __restrict__ out) {
    int wave = threadIdx.x >> 5;
    int lane = threadIdx.x & 31;
    int tile = blockIdx.x * WAVES_PER_BLOCK + wave;
    int r0 = (tile / TILES_X) * 16;
    int c0 = (tile % TILES_X) * 16;
    int half = lane >> 4;
    int lm   = lane & 15;
    long rowoff = (long)(r0 + lm) * IMG_W;

    v8f acc = {};
#pragma unroll
    for (int ch = 0; ch < 6; ++ch) {
        int k0  = ch * 4 + 2 * half;         // K indices {k0, k0+1} for A and B
        int col = c0 - 4 + k0;               // even
        bool ok = (col >= 0) & (col < IMG_W);
        int colc = ok ? col : 0;             // clamped -> load always legal
        v2f a = *(const v2f*)(x + rowoff + colc);   // global_load_b64, 8B aligned
        a[0] = ok ? a[0] : 0.0f;
        a[1] = ok ? a[1] : 0.0f;
        v2f b;
        b[0] = (k0     >= lm && k0     <= lm + 8) ? 1.0f : 0.0f;
        b[1] = (k0 + 1 >= lm && k0 + 1 <= lm + 8) ? 1.0f : 0.0f;
        acc = __builtin_amdgcn_wmma_f32_16x16x4_f32(false, a, false, b,
                                                    (short)0, acc, false, false);
    }
#pragma unroll
    for (int v = 0; v < 8; ++v)
        out[(r0 + v + 8 * half) * IMG_W + c0 + lm] = acc[v];
}

// ---------------------------------------------------------------------------
// 2) Vertical 9-tap pass (x 1/81) via WMMA: O = Band(16x24) x Hsum(24x16)
//    B loads are lane-coalesced (lane = column); row clamp + select.
// ---------------------------------------------------------------------------
__global__ __launch_bounds__(256) void blur_v_wmma(const float* __restrict__ hsum,
                                                   float* __restrict__ out) {
    int wave = threadIdx.x >> 5;
    int lane = threadIdx.x & 31;
    int tile = blockIdx.x * WAVES_PER_BLOCK + wave;
    int r0 = (tile / TILES_X) * 16;
    int c0 = (tile % TILES_X) * 16;
    int half = lane >> 4;
    int lm   = lane & 15;
    const float w = 1.0f / 81.0f;

    v8f acc = {};
#pragma unroll
    for (int ch = 0; ch < 6; ++ch) {
        int k0 = ch * 4 + 2 * half;
        v2f a;
        a[0] = (k0     >= lm && k0     <= lm + 8) ? w : 0.0f;
        a[1] = (k0 + 1 >= lm && k0 + 1 <= lm + 8) ? w : 0.0f;

        int rA = r0 + k0 - 4;
        int rB = r0 + k0 - 3;
        bool okA = (rA >= 0) & (rA < IMG_H);
        bool okB = (rB >= 0) & (rB < IMG_H);
        int rAc = okA ? rA : 0;
        int rBc = okB ? rB : 0;
        v2f b;
        float t0 = hsum[(long)rAc * IMG_W + c0 + lm];
        float t1 = hsum[(long)rBc * IMG_W + c0 + lm];
        b[0] = okA ? t0 : 0.0f;
        b[1] = okB ? t1 : 0.0f;
        acc = __builtin_amdgcn_wmma_f32_16x16x4_f32(false, a, false, b,
                                                    (short)0, acc, false, false);
    }
#pragma unroll
    for (int v = 0; v < 8; ++v)
        out[(r0 + v + 8 * half) * IMG_W + c0 + lm] = acc[v];
}

// ---------------------------------------------------------------------------
// 3) Per-patch exact order statistics via bitwise binary search.
//    v1 = K1-th largest, v2 = K2-th largest of the 65536 blurred values.
//    frac_above(t) >= L  <=>  count(x>t) >= K  <=>  t < v_K.
//    One block (1024 threads) per patch; patch cached in registers via b128.
// ---------------------------------------------------------------------------
__global__ __launch_bounds__(1024) void patch_quantiles(const float* __restrict__ blur,
                                                        float* __restrict__ qv) {
    int p  = blockIdx.x;        // 0..63
    int pr = p >> 3, pc = p & 7;
    const float* base = blur + (pr * 256) * IMG_W + pc * 256;

    float vals[64];
#pragma unroll
    for (int i = 0; i < 16; ++i) {
        int e = i * 4096 + (int)threadIdx.x * 4;  // 0..65535, 16B aligned
        v4f q = *(const v4f*)(base + (e >> 8) * IMG_W + (e & 255));
        vals[i * 4 + 0] = q[0];
        vals[i * 4 + 1] = q[1];
        vals[i * 4 + 2] = q[2];
        vals[i * 4 + 3] = q[3];
    }

    bool frame = (pr == 0) || (pr == 7) || (pc == 0) || (pc == 7);
    float lo = frame ? (0.47f - 0.05f) : 0.47f;   // PATCH_MEAN+0.02-f_bias (fp32)
    float hi = 0.43f;                              // PATCH_MEAN-0.02
    int K1 = (int)__builtin_ceil((double)lo * 65536.0);       // frac>=lo <=> cnt>=K1
    int K2 = (int)__builtin_floor((double)hi * 65536.0) + 1;  // frac> hi <=> cnt>=K2

    __shared__ int scnt;
    for (int which = 0; which < 2; ++which) {
        int K = which ? K2 : K1;
        unsigned ulo = 0u;             // t=0: all blurred values > 0 -> cnt >= K
        unsigned uhi = 0x7F800000u;    // t=+inf: cnt=0 < K
        while (uhi - ulo > 1u) {
            unsigned mid = (ulo + uhi) >> 1;
            float tv = __uint_as_float(mid);
            int c = 0;
#pragma unroll
            for (int i = 0; i < 64; ++i) c += (vals[i] > tv) ? 1 : 0;
            for (int off = 16; off > 0; off >>= 1) c += __shfl_down(c, off, 32);
            if (threadIdx.x == 0) scnt = 0;
            __syncthreads();
            if ((threadIdx.x & 31) == 0) atomicAdd(&scnt, c);
            __syncthreads();
            int total = scnt;
            __syncthreads();
            if (total >= K) ulo = mid; else uhi = mid;
        }
        // uhi = smallest t with count(x>t) < K  ==  K-th largest value
        if (threadIdx.x == 0) qv[which * 64 + p] = __uint_as_float(uhi);
    }
}

// ---------------------------------------------------------------------------
// 4) Bit-exact replay of the sequential fp32 threshold walk across patches.
//    frac<lo <=> cnt<K1 <=> th>=v1 ; frac>hi <=> cnt>=K2 <=> th<v2
// ---------------------------------------------------------------------------
__global__ void th_scan(const float* __restrict__ qv, float* __restrict__ ths) {
    if (threadIdx.x != 0 || blockIdx.x != 0) return;
    float th = 0.5f;
    for (int p = 0; p < 64; ++p) {
        float v1 = qv[p];
        float v2 = qv[64 + p];
        while (th >= v1) th -= 5e-5f;
        while (th < v2)  th += 5e-6f;
        ths[p] = th;
    }
}

// ---------------------------------------------------------------------------
// 5) Binarize (per-source-pixel patch threshold) fused with horizontal max.
//    Branch-free: clamp coordinate, select contribution to -inf when invalid.
// ---------------------------------------------------------------------------
__global__ __launch_bounds__(256) void bin_rowmax(const float* __restrict__ blur,
                                                  const float* __restrict__ ths,
                                                  float* __restrict__ out) {
    int idx = blockIdx.x * 256 + threadIdx.x;
    int r = idx >> 11, c = idx & (IMG_W - 1);
    int pr8 = (r >> 8) << 3;
    long rowoff = (long)r * IMG_W;
    float m = -__builtin_inff();
#pragma unroll
    for (int dx = -4; dx <= 4; ++dx) {
        int cc = c + dx;
        bool ok = (cc >= 0) & (cc < IMG_W);
        int ccc = ok ? cc : 0;
        float v  = blur[rowoff + ccc];
        float th = ths[pr8 + (ccc >> 8)];
        float b  = (v > th) ? 1.0f : 0.0f;
        b = ok ? b : -__builtin_inff();
        m = fmaxf(m, b);
    }
    out[idx] = m;
}

// OP: 0 = max, 1 = min.  DIR: 0 = along columns (vertical), 1 = along rows.
template <int OP, int DIR>
__global__ __launch_bounds__(256) void pool_pass(const float* __restrict__ in,
                                                 float* __restrict__ out) {
    int idx = blockIdx.x * 256 + threadIdx.x;
    int r = idx >> 11, c = idx & (IMG_W - 1);
    const float ident = OP ? __builtin_inff() : -__builtin_inff();
    float m = ident;
#pragma unroll
    for (int d = -4; d <= 4; ++d) {
        int rr = DIR ? r : r + d;
        int cc = DIR ? c + d : c;
        bool ok = (rr >= 0) & (rr < IMG_H) & (cc >= 0) & (cc < IMG_W);
        int rrc = ok ? rr : 0;
        int ccc = ok ? cc : 0;
        float v = in[(long)rrc * IMG_W + ccc];
        v = ok ? v : ident;
        m = OP ? fminf(m, v) : fmaxf(m, v);
    }
    out[idx] = m;
}

// ---------------------------------------------------------------------------
extern "C" void kernel_launch(void* const* d_in, const int* in_sizes, int n_in,
                              void* d_out, int out_size, void* d_ws, size_t ws_size,
                              hipStream_t stream) {
    const float* x = (const float*)d_in[0];        // (1,1,2048,2048) f32
    float* out  = (float*)d_out;                   // also used as ping buffer
    float* buf0 = (float*)d_ws;                    // 2048*2048 f32 scratch
    float* qv   = buf0 + IMG_W * IMG_H;            // v1[64] ++ v2[64]
    float* ths  = qv + 128;                        // 64 thresholds

    const int tile_blocks = NTILES / WAVES_PER_BLOCK;   // 2048
    const int pix_blocks  = (IMG_W * IMG_H) / 256;      // 16384

    // blurred ends up in d_out (temporarily)
    blur_h_wmma<<<tile_blocks, 256, 0, stream>>>(x, buf0);
    blur_v_wmma<<<tile_blocks, 256, 0, stream>>>(buf0, out);

    patch_quantiles<<<64, 1024, 0, stream>>>(out, qv);
    th_scan<<<1, 32, 0, stream>>>(qv, ths);

    // maxpool = colmax(rowmax(bin)); minpool = colmin(rowmin(maxpooled))
    bin_rowmax<<<pix_blocks, 256, 0, stream>>>(out, ths, buf0);
    pool_pass<0, 0><<<pix_blocks, 256, 0, stream>>>(buf0, out);   // vertical max
    pool_pass<1, 1><<<pix_blocks, 256, 0, stream>>>(out, buf0);   // horizontal min
    pool_pass<1, 0><<<pix_blocks, 256, 0, stream>>>(buf0, out);   // vertical min
}